// RoIAlignMax_43241730736874
// MI455X (gfx1250) — compile-verified
//
#include <hip/hip_runtime.h>

#define SPATIAL_SCALE 0.0625f

namespace {
constexpr int kC      = 512;
constexpr int kH      = 64;
constexpr int kW      = 64;
constexpr int kPlane  = kH * kW;          // 4096 floats per (b,c) plane
constexpr int kChunk  = 64;               // channels staged per TDM chunk
constexpr int kNChunk = kC / kChunk;      // 8 chunks per RoI
constexpr int kTileW  = 24;               // staged cols (floats), 96B rows (16B aligned)
constexpr int kTileH  = 18;               // staged rows (max bilinear footprint)
constexpr int kChStr  = kTileW * kTileH;  // 432 floats per channel in LDS
constexpr int kBufF   = kChunk * kChStr;  // 27648 floats per buffer (~108 KB)
constexpr int kThreads = 512;             // 16 wave32 waves
}

#if defined(__AMDGCN__) && __has_builtin(__builtin_amdgcn_tensor_load_to_lds) && \
    __has_builtin(__builtin_amdgcn_s_wait_tensorcnt)
#define USE_TDM 1
#else
#define USE_TDM 0
#endif

typedef __attribute__((ext_vector_type(4))) unsigned int v4u;
typedef __attribute__((ext_vector_type(4))) int v4i;
typedef __attribute__((ext_vector_type(8))) int v8i;

#if USE_TDM
__device__ __forceinline__ int rfl(int x) { return __builtin_amdgcn_readfirstlane(x); }

__device__ __forceinline__ unsigned int lds_byte_offset(const void* p) {
  typedef __attribute__((address_space(3))) const void* lds_cptr;
  return (unsigned int)(__SIZE_TYPE__)(lds_cptr)p;
}

// Issue one TDM load: 3D tile (x=kTileW f32, y=kTileH rows, z=kChunk channels)
// from global (strides 64 / 4096 floats) into a packed LDS tile.
// tensor_dim0/1 are set to the remaining extent so TDM zero-fills past the
// image edge instead of reading out of bounds.
__device__ __forceinline__ void tdm_issue(const float* gsrc, unsigned int lds_off,
                                          int cols_rem, int rows_rem) {
  unsigned long long ga = (unsigned long long)(__SIZE_TYPE__)gsrc;
  const unsigned int ga_lo = (unsigned int)rfl((int)(unsigned int)ga);
  const unsigned int ga_hi = (unsigned int)rfl((int)(unsigned int)(ga >> 32));
  const unsigned int ldso  = (unsigned int)rfl((int)lds_off);
  const unsigned long long td0 = (unsigned int)rfl(cols_rem);   // tensor_dim0
  const unsigned long long td1 = (unsigned int)rfl(rows_rem);   // tensor_dim1

  v4u g0;
  g0.x = 1u;                                        // count=1 valid descriptor
  g0.y = ldso;                                      // lds_addr (bytes)
  g0.z = ga_lo;                                     // global_addr[31:0]
  g0.w = (ga_hi & 0x01FFFFFFu) | 0x80000000u;       // global_addr[56:32] | type=2

  // Group1 (256b): [17:16]=data_size(4B) [79:48]=td0 [111:80]=td1
  // [127:112]=tile0 [143:128]=tile1 [159:144]=tile2 [207:160]=stride0 [255:208]=stride1
  const unsigned long long q0 = (2ull << 16) | (td0 << 48);
  const unsigned long long q1 = (td0 >> 16) | (td1 << 16) |
                                ((unsigned long long)kTileW << 48);
  const unsigned long long q2 = (unsigned long long)kTileH |
                                ((unsigned long long)kChunk << 16) |
                                ((unsigned long long)kW << 32);
  const unsigned long long q3 = ((unsigned long long)kPlane << 16);

  v8i g1;
  g1[0] = (int)(unsigned int)q0;  g1[1] = (int)(unsigned int)(q0 >> 32);
  g1[2] = (int)(unsigned int)q1;  g1[3] = (int)(unsigned int)(q1 >> 32);
  g1[4] = (int)(unsigned int)q2;  g1[5] = (int)(unsigned int)(q2 >> 32);
  g1[6] = (int)(unsigned int)q3;  g1[7] = (int)(unsigned int)(q3 >> 32);

  v4i g2;
  g2.x = kChunk;   // tensor_dim2 (>= tile_dim2, never OOB in z)
  g2.y = 0;        // tensor_dim3 (unused)
  g2.z = kPlane;   // tensor_dim2_stride low 32 (4096 floats per channel)
  g2.w = 0;        // stride hi | tile_dim3=0

  v4i g3 = {0, 0, 0, 0};

#if __clang_major__ >= 23
  // clang-23 / therock lane: 6-arg form with a trailing reserved int32x8
  // (ISA: VADDR4 unused -> NULL/zeros).
  v8i g4 = {0, 0, 0, 0, 0, 0, 0, 0};
  __builtin_amdgcn_tensor_load_to_lds(g0, g1, g2, g3, g4, 0);
#else
  // ROCm 7.2 / clang-22 lane: 5-arg form.
  __builtin_amdgcn_tensor_load_to_lds(g0, g1, g2, g3, 0);
#endif
}
#endif

__global__ __launch_bounds__(kThreads) void roialign_maxpool_kernel(
    const float* __restrict__ feat, const float* __restrict__ rois,
    float* __restrict__ out) {
  extern __shared__ float smem[];   // 2 * kBufF floats (double buffer)

  const int n = (int)blockIdx.x;
  const float* roi = rois + 5 * n;
  const int   b  = (int)roi[0];
  const float x1 = roi[1] * SPATIAL_SCALE;
  const float y1 = roi[2] * SPATIAL_SCALE;
  const float x2 = roi[3] * SPATIAL_SCALE;
  const float y2 = roi[4] * SPATIAL_SCALE;
  const float bw = fmaxf(x2 - x1, 0.0f) * (1.0f / 7.0f);
  const float bh = fmaxf(y2 - y1, 0.0f) * (1.0f / 7.0f);

  int row_base = (int)floorf(y1);
  row_base = min(max(row_base, 0), kH - 1);
  int col_base = ((int)floorf(x1)) & ~3;            // 16B-aligned column origin
  col_base = min(max(col_base, 0), kW - 4);
  const int rows_rem = kH - row_base;
  const int cols_rem = kW - col_base;

  const int tid = (int)threadIdx.x;
  const int cl  = tid >> 3;        // channel within chunk: 0..63
  const int gi  = tid & 7;         // output row: 0..6 active, 7 copy-only
  const bool active = gi < 7;

  // ---- column taps (identical for all threads, kept in registers) ----
  int   c0[8], c1[8];
  float wc0[8], wc1[8];
#pragma unroll
  for (int j = 0; j < 8; ++j) {
    const float w  = fmaf((float)j, bw, x1);
    const int   w0 = (int)floorf(w);
    const float wr = w - (float)w0;
    float a0 = 1.0f - wr, a1 = wr;
    if (!(w >= 0.0f && w < (float)kW)) { a0 = 0.0f; a1 = 0.0f; }  // inside mask
    const int i0 = min(max(w0, 0), kW - 1) - col_base;
    const int i1 = min(max(w0 + 1, 0), kW - 1) - col_base;
    c0[j]  = min(max(i0, 0), kTileW - 1);
    c1[j]  = min(max(i1, 0), kTileW - 1);
    wc0[j] = a0; wc1[j] = a1;
  }

  // ---- row taps for sample rows gi and gi+1 ----
  int rA0, rA1, rB0, rB1;
  float whA0, whA1, whB0, whB1;
  {
    const float hA = fmaf((float)gi, bh, y1);
    int h0 = (int)floorf(hA);
    float hr = hA - (float)h0;
    whA0 = 1.0f - hr; whA1 = hr;
    if (!(hA >= 0.0f && hA < (float)kH)) { whA0 = 0.0f; whA1 = 0.0f; }
    rA0 = min(max(min(max(h0, 0), kH - 1) - row_base, 0), kTileH - 1);
    rA1 = min(max(min(max(h0 + 1, 0), kH - 1) - row_base, 0), kTileH - 1);

    const float hB = fmaf((float)(gi + 1), bh, y1);
    h0 = (int)floorf(hB);
    hr = hB - (float)h0;
    whB0 = 1.0f - hr; whB1 = hr;
    if (!(hB >= 0.0f && hB < (float)kH)) { whB0 = 0.0f; whB1 = 0.0f; }
    rB0 = min(max(min(max(h0, 0), kH - 1) - row_base, 0), kTileH - 1);
    rB1 = min(max(min(max(h0 + 1, 0), kH - 1) - row_base, 0), kTileH - 1);
  }

  const float* gchunk0 = feat + (__SIZE_TYPE__)b * kC * kPlane
                       + row_base * kW + col_base;

#if USE_TDM
  if (tid == 0) {
    tdm_issue(gchunk0, lds_byte_offset(smem), cols_rem, rows_rem);
  }
#endif

  for (int k = 0; k < kNChunk; ++k) {
    float* buf = smem + (k & 1) * kBufF;

#if USE_TDM
    if (tid == 0) {
      if (k + 1 < kNChunk) {
        // Prefetch next chunk's DMA, then wait only for the current one
        // (TDM ops complete in order per wave -> tensorcnt<=1 means chunk k done).
        tdm_issue(gchunk0 + (__SIZE_TYPE__)(k + 1) * kChunk * kPlane,
                  lds_byte_offset(smem + ((k + 1) & 1) * kBufF),
                  cols_rem, rows_rem);
        __builtin_amdgcn_s_wait_tensorcnt(1);
      } else {
        __builtin_amdgcn_s_wait_tensorcnt(0);
      }
    }
#else
    // Synchronous cooperative staging (fallback if the TDM builtin is absent).
    {
      const float* gsrc = gchunk0 + (__SIZE_TYPE__)k * kChunk * kPlane;
      constexpr int kSegs  = kTileW / 4;             // 6 float4 per row
      constexpr int kUnits = kChunk * kTileH * kSegs;
      for (int u = tid; u < kUnits; u += kThreads) {
        const int c  = u / (kTileH * kSegs);
        const int r  = (u / kSegs) % kTileH;
        const int sg = u % kSegs;
        if (r < rows_rem && sg * 4 < cols_rem) {
          const float4 v = *(const float4*)(gsrc + c * kPlane + r * kW + sg * 4);
          *(float4*)(buf + c * kChStr + r * kTileW + sg * 4) = v;
        }
      }
    }
#endif
    __syncthreads();

    if (active) {
      const float* Lc  = buf + cl * kChStr;
      const float* pA0 = Lc + rA0 * kTileW;
      const float* pA1 = Lc + rA1 * kTileW;
      const float* pB0 = Lc + rB0 * kTileW;
      const float* pB1 = Lc + rB1 * kTileW;
      float A[8], Bv[8];
#pragma unroll
      for (int j = 0; j < 8; ++j) {
        const float ta = wc0[j] * pA0[c0[j]] + wc1[j] * pA0[c1[j]];
        const float tb = wc0[j] * pA1[c0[j]] + wc1[j] * pA1[c1[j]];
        A[j] = whA0 * ta + whA1 * tb;
        const float tc = wc0[j] * pB0[c0[j]] + wc1[j] * pB0[c1[j]];
        const float td = wc0[j] * pB1[c0[j]] + wc1[j] * pB1[c1[j]];
        Bv[j] = whB0 * tc + whB1 * td;
      }
      const int c = k * kChunk + cl;
      float* o = out + ((__SIZE_TYPE__)n * kC + c) * 49 + gi * 7;
#pragma unroll
      for (int jj = 0; jj < 7; ++jj) {
        o[jj] = fmaxf(fmaxf(A[jj], A[jj + 1]), fmaxf(Bv[jj], Bv[jj + 1]));
      }
    }
    __syncthreads();   // buffer (k&1) may be overwritten by DMA issued next iter
  }
}

extern "C" void kernel_launch(void* const* d_in, const int* in_sizes, int n_in,
                              void* d_out, int out_size, void* d_ws, size_t ws_size,
                              hipStream_t stream) {
  const float* feat = (const float*)d_in[0];
  const float* rois = (const float*)d_in[1];
  float* out = (float*)d_out;
  const int n_rois = in_sizes[1] / 5;                 // 2000
  const size_t shmem = (size_t)2 * kBufF * sizeof(float);  // ~216 KB of the 320 KB WGP LDS
  roialign_maxpool_kernel<<<n_rois, kThreads, shmem, stream>>>(feat, rois, out);
}